// LSTMClassifier_63505386438888
// MI455X (gfx1250) — compile-verified
//
#include <hip/hip_runtime.h>
#include <hip/hip_bf16.h>
#include <math.h>

// ---------------- problem constants ----------------
#define Bb   64
#define Tt   512
#define INd  1024
#define HID  256
#define G4   1024   // 4*HID
#define NCc  8
// persistent kernel: 8 WGs x 1024 thr = 256 waves = 64 groups x 4 K-splits
#define PWG  8
#define PTHR 1024
#define GRP_PER_WG 8

typedef __attribute__((ext_vector_type(2))) float v2f;
typedef __attribute__((ext_vector_type(8))) float v8f;

// Full-precision CDNA5 WMMA: D(16x16 f32) = A(16x4 f32) x B(4x16 f32) + C
__device__ __forceinline__ v8f wmma4(v2f a, v2f b, v8f c) {
  return __builtin_amdgcn_wmma_f32_16x16x4_f32(false, a, false, b, (short)0, c,
                                               false, false);
}

__device__ __forceinline__ float sigm(float x) { return 1.0f / (1.0f + __expf(-x)); }

// 4-accumulator K-panel: one A frag feeds 4 B frags (4 N-tiles / 4 gates).
// ar    : &A[m0+ml][k0+koff]            (A row-major, row stride = K row len)
// wbase : &W[n_tile0 + ml][k0+koff]     (B = W^T, W row-major)
// wqstride: element offset between successive N-tiles' W rows
__device__ __forceinline__ void gemm4(const float* __restrict__ ar,
                                      const float* __restrict__ wbase,
                                      size_t wqstride, int klen, v8f acc[4]) {
#pragma unroll 4
  for (int k = 0; k < klen; k += 4) {
    v2f a = *(const v2f*)(ar + k);
#pragma unroll
    for (int q = 0; q < 4; ++q) {
      v2f b = *(const v2f*)(wbase + q * wqstride + k);
      acc[q] = wmma4(a, b, acc[q]);
    }
  }
}

// Sense-reversing grid barrier over PWG co-resident workgroups.
__device__ __forceinline__ void grid_sync(int* cnt, int* gen, int& lg) {
  __threadfence();
  __syncthreads();
  if (threadIdx.x == 0) {
    int prev = __hip_atomic_fetch_add(cnt, 1, __ATOMIC_ACQ_REL,
                                      __HIP_MEMORY_SCOPE_AGENT);
    if (prev == PWG - 1) {
      __hip_atomic_store(cnt, 0, __ATOMIC_RELAXED, __HIP_MEMORY_SCOPE_AGENT);
      __hip_atomic_fetch_add(gen, 1, __ATOMIC_ACQ_REL, __HIP_MEMORY_SCOPE_AGENT);
    } else {
      while (__hip_atomic_load(gen, __ATOMIC_ACQUIRE,
                               __HIP_MEMORY_SCOPE_AGENT) <= lg) {
        __builtin_amdgcn_s_sleep(1);
      }
    }
    lg++;
  }
  __syncthreads();
  __threadfence();
}

__global__ void init_sync_kernel(int* scnt, int* sgen) {
  if (threadIdx.x == 0) { *scnt = 0; *sgen = 0; }
}

// Combine 4 K-split partials through LDS, then wave kq==0 applies the LSTM
// cell in registers and writes h/c. acc[0..3] = i,f,g,o partial gate tiles.
__device__ __forceinline__ void reduce_act(v8f acc[4], float* gbuf, int kq,
                                           int mrow, int n0, int ml, int half,
                                           int t, float* __restrict__ cbuf,
                                           float* __restrict__ hbuf) {
  if (kq == 1) {
#pragma unroll
    for (int q = 0; q < 4; ++q)
#pragma unroll
      for (int j = 0; j < 8; ++j)
        gbuf[q * 256 + (j + 8 * half) * 16 + ml] = acc[q][j];
  }
  __syncthreads();
  if (kq >= 2) {
#pragma unroll
    for (int q = 0; q < 4; ++q)
#pragma unroll
      for (int j = 0; j < 8; ++j)
        atomicAdd(&gbuf[q * 256 + (j + 8 * half) * 16 + ml], acc[q][j]);
  }
  __syncthreads();
  if (kq == 0) {
#pragma unroll
    for (int j = 0; j < 8; ++j) {
      const int idxr = (j + 8 * half) * 16 + ml;
      const float ig = sigm(acc[0][j] + gbuf[idxr]);
      const float fg = sigm(acc[1][j] + gbuf[256 + idxr]);
      const float gg = tanhf(acc[2][j] + gbuf[512 + idxr]);
      const float og = sigm(acc[3][j] + gbuf[768 + idxr]);
      const int ci = (mrow + j) * HID + n0 + ml;
      const float cp = (t == 0) ? 0.0f : cbuf[ci];
      const float cn = fg * cp + ig * gg;
      cbuf[ci] = cn;
      hbuf[ci] = og * tanhf(cn);
    }
  }
}

// ---------------------------------------------------------------------------
// Kernel 1: XW0[b*T+t][n] = x[b][t][:] . W_ih0[n][:] + b_ih0[n] + b_hh0[n]
// M=32768, N=1024, K=1024; 4 N-tiles per wave (A-frag reuse x4).
// ---------------------------------------------------------------------------
__global__ void __launch_bounds__(256)
xw0_kernel(const float* __restrict__ x, const float* __restrict__ w,
           const float* __restrict__ bih, const float* __restrict__ bhh,
           float* __restrict__ xw0) {
  const int tid  = threadIdx.x;
  const int wid  = blockIdx.x * 8 + (tid >> 5);           // 0..32767
  const int lane = tid & 31, ml = lane & 15, half = lane >> 4, koff = 2 * half;
  const int mt = wid >> 4, nt4 = wid & 15;                // 2048 x 16
  const int m0 = mt * 16, n0g = nt4 * 64;                 // 4 tiles: n0g+16q

  v8f acc[4];
#pragma unroll
  for (int q = 0; q < 4; ++q) {
    const float bv = bih[n0g + 16 * q + ml] + bhh[n0g + 16 * q + ml];
#pragma unroll
    for (int j = 0; j < 8; ++j) acc[q][j] = bv;
  }
  gemm4(x + (size_t)(m0 + ml) * INd + koff,
        w + (size_t)(n0g + ml) * INd + koff,
        (size_t)16 * INd, INd, acc);
#pragma unroll
  for (int q = 0; q < 4; ++q)
#pragma unroll
    for (int j = 0; j < 8; ++j)
      xw0[(size_t)(m0 + 8 * half + j) * G4 + n0g + 16 * q + ml] = acc[q][j];
}

// ---------------------------------------------------------------------------
// Kernel 2: persistent sequential LSTM, 3 layers x 512 steps.
// 64 groups (4 Mtiles x 16 HID-column tiles) x 4 K-split waves.
// Each group owns all 4 gates of its tile -> activation fused in registers.
// 3 grid barriers per step; gates never touch global memory.
// ---------------------------------------------------------------------------
__global__ void __launch_bounds__(PTHR)
lstm_seq_kernel(const float* __restrict__ xw0,
                const float* __restrict__ Whh0,
                const float* __restrict__ Wih1, const float* __restrict__ Whh1,
                const float* __restrict__ bih1, const float* __restrict__ bhh1,
                const float* __restrict__ Wih2, const float* __restrict__ Whh2,
                const float* __restrict__ bih2, const float* __restrict__ bhh2,
                float* __restrict__ h0, float* __restrict__ h1,
                float* __restrict__ c0, float* __restrict__ c1,
                float* __restrict__ c2,
                float* __restrict__ h2all,
                int* __restrict__ scnt, int* __restrict__ sgen) {
  __shared__ float red[GRP_PER_WG][4 * 256];               // 32 KB

  const int tid  = threadIdx.x;
  const int lw   = tid >> 5;                               // wave in WG, 0..31
  const int lane = tid & 31, ml = lane & 15, half = lane >> 4, koff = 2 * half;
  const int kq   = lw & 3;                                 // K-split id
  const int gw   = lw >> 2;                                // group in WG, 0..7
  const int g    = blockIdx.x * GRP_PER_WG + gw;           // 0..63
  const int mt   = g >> 4, ntH = g & 15;                   // 4 x 16
  const int m0   = mt * 16, n0 = ntH * 16;
  const int mrow = m0 + 8 * half;
  float* gbuf = &red[gw][0];
  const int arowoff = (m0 + ml) * HID + koff;
  const size_t wqs = (size_t)HID * HID;                    // gate stride in W
  int lg = 0;

  float bias1[4], bias2[4];
#pragma unroll
  for (int q = 0; q < 4; ++q) {
    bias1[q] = bih1[q * HID + n0 + ml] + bhh1[q * HID + n0 + ml];
    bias2[q] = bih2[q * HID + n0 + ml] + bhh2[q * HID + n0 + ml];
  }

  for (int t = 0; t < Tt; ++t) {
    // ---------------- layer 0: XW0[:,t,:] + h0 @ Whh0^T ----------------
    {
      v8f acc[4];
      if (kq == 0) {
#pragma unroll
        for (int q = 0; q < 4; ++q)
#pragma unroll
          for (int j = 0; j < 8; ++j)
            acc[q][j] =
                xw0[((size_t)(mrow + j) * Tt + t) * G4 + q * HID + n0 + ml];
      } else {
#pragma unroll
        for (int q = 0; q < 4; ++q)
#pragma unroll
          for (int j = 0; j < 8; ++j) acc[q][j] = 0.0f;
      }
      if (t > 0) {
        const int k0 = kq * 64;                            // K=256 / 4 waves
        gemm4(h0 + arowoff + k0,
              Whh0 + (size_t)(n0 + ml) * HID + k0 + koff, wqs, 64, acc);
      }
      reduce_act(acc, gbuf, kq, mrow, n0, ml, half, t, c0, h0);
    }
    grid_sync(scnt, sgen, lg);
    // ---------------- layer 1: h0 @ Wih1^T + h1 @ Whh1^T ----------------
    {
      v8f acc[4];
#pragma unroll
      for (int q = 0; q < 4; ++q) {
        const float bv = (kq == 0) ? bias1[q] : 0.0f;
#pragma unroll
        for (int j = 0; j < 8; ++j) acc[q][j] = bv;
      }
      if (kq < 2) {                                        // ih half, K=256/2
        const int k0 = kq * 128;
        gemm4(h0 + arowoff + k0,
              Wih1 + (size_t)(n0 + ml) * HID + k0 + koff, wqs, 128, acc);
      } else if (t > 0) {                                  // hh half
        const int k0 = (kq - 2) * 128;
        gemm4(h1 + arowoff + k0,
              Whh1 + (size_t)(n0 + ml) * HID + k0 + koff, wqs, 128, acc);
      }
      reduce_act(acc, gbuf, kq, mrow, n0, ml, half, t, c1, h1);
    }
    grid_sync(scnt, sgen, lg);
    // ---------------- layer 2: h1 @ Wih2^T + h2 @ Whh2^T ----------------
    {
      v8f acc[4];
#pragma unroll
      for (int q = 0; q < 4; ++q) {
        const float bv = (kq == 0) ? bias2[q] : 0.0f;
#pragma unroll
        for (int j = 0; j < 8; ++j) acc[q][j] = bv;
      }
      if (kq < 2) {
        const int k0 = kq * 128;
        gemm4(h1 + arowoff + k0,
              Wih2 + (size_t)(n0 + ml) * HID + k0 + koff, wqs, 128, acc);
      } else if (t > 0) {
        const int k0 = (kq - 2) * 128;
        gemm4(h2all + (size_t)(t - 1) * Bb * HID + arowoff + k0,
              Whh2 + (size_t)(n0 + ml) * HID + k0 + koff, wqs, 128, acc);
      }
      reduce_act(acc, gbuf, kq, mrow, n0, ml, half, t, c2,
                 h2all + (size_t)t * Bb * HID);
    }
    grid_sync(scnt, sgen, lg);
  }
}

// ---------------------------------------------------------------------------
// Kernel 3: R1[t*64+b][n] = relu(h2all . fc1_w^T + fc1_b)
// M=32768, N=128, K=256; 4 N-tiles per wave.
// ---------------------------------------------------------------------------
__global__ void __launch_bounds__(256)
fc1_kernel(const float* __restrict__ h2all, const float* __restrict__ w,
           const float* __restrict__ bias, float* __restrict__ r1) {
  const int tid  = threadIdx.x;
  const int wid  = blockIdx.x * 8 + (tid >> 5);           // 0..4095
  const int lane = tid & 31, ml = lane & 15, half = lane >> 4, koff = 2 * half;
  const int mt = wid >> 1, ng = wid & 1;                  // 2048 x 2
  const int m0 = mt * 16, n0 = ng * 64;

  v8f acc[4];
#pragma unroll
  for (int q = 0; q < 4; ++q) {
    const float bv = bias[n0 + 16 * q + ml];
#pragma unroll
    for (int j = 0; j < 8; ++j) acc[q][j] = bv;
  }
  gemm4(h2all + (size_t)(m0 + ml) * HID + koff,
        w + (size_t)(n0 + ml) * HID + koff,
        (size_t)16 * HID, HID, acc);
#pragma unroll
  for (int q = 0; q < 4; ++q)
#pragma unroll
    for (int j = 0; j < 8; ++j)
      r1[(size_t)(m0 + 8 * half + j) * 128 + n0 + 16 * q + ml] =
          fmaxf(acc[q][j], 0.0f);
}

// ---------------------------------------------------------------------------
// Kernel 4: out[b][nc] = fc2_b[nc] + (1/T) * sum_t R1[t*64+b][:] . fc2_w[nc][:]
// ---------------------------------------------------------------------------
__global__ void __launch_bounds__(256)
fc2_kernel(const float* __restrict__ r1, const float* __restrict__ w2,
           const float* __restrict__ b2, float* __restrict__ out) {
  __shared__ float sw[NCc * 128];
  __shared__ float red[256 * NCc];
  const int b = blockIdx.x, tid = threadIdx.x;
  for (int i = tid; i < NCc * 128; i += 256) sw[i] = w2[i];
  __syncthreads();

  float acc[NCc];
#pragma unroll
  for (int n = 0; n < NCc; ++n) acc[n] = 0.0f;
  for (int t = tid; t < Tt; t += 256) {
    const float* r = r1 + (size_t)(t * Bb + b) * 128;
    for (int k = 0; k < 128; ++k) {
      const float v = r[k];
#pragma unroll
      for (int n = 0; n < NCc; ++n) acc[n] += v * sw[n * 128 + k];
    }
  }
#pragma unroll
  for (int n = 0; n < NCc; ++n) red[tid * NCc + n] = acc[n];
  __syncthreads();
  if (tid < NCc) {
    float s = 0.0f;
    for (int i = 0; i < 256; ++i) s += red[i * NCc + tid];
    out[b * NCc + tid] = b2[tid] + s * (1.0f / (float)Tt);
  }
}

// ---------------------------------------------------------------------------
extern "C" void kernel_launch(void* const* d_in, const int* in_sizes, int n_in,
                              void* d_out, int out_size, void* d_ws,
                              size_t ws_size, hipStream_t stream) {
  const float* x    = (const float*)d_in[0];
  const float* Wih0 = (const float*)d_in[1];
  const float* Whh0 = (const float*)d_in[2];
  const float* bih0 = (const float*)d_in[3];
  const float* bhh0 = (const float*)d_in[4];
  const float* Wih1 = (const float*)d_in[5];
  const float* Whh1 = (const float*)d_in[6];
  const float* bih1 = (const float*)d_in[7];
  const float* bhh1 = (const float*)d_in[8];
  const float* Wih2 = (const float*)d_in[9];
  const float* Whh2 = (const float*)d_in[10];
  const float* bih2 = (const float*)d_in[11];
  const float* bhh2 = (const float*)d_in[12];
  const float* fc1w = (const float*)d_in[13];
  const float* fc1b = (const float*)d_in[14];
  const float* fc2w = (const float*)d_in[15];
  const float* fc2b = (const float*)d_in[16];

  float* ws = (float*)d_ws;
  size_t off = 0;
  float* xw0   = ws + off; off += (size_t)Bb * Tt * G4;    // 128 MB
  float* h2all = ws + off; off += (size_t)Tt * Bb * HID;   //  32 MB
  float* r1    = ws + off; off += (size_t)Tt * Bb * 128;   //  16 MB
  float* h0 = ws + off; off += (size_t)Bb * HID;
  float* h1 = ws + off; off += (size_t)Bb * HID;
  float* c0 = ws + off; off += (size_t)Bb * HID;
  float* c1 = ws + off; off += (size_t)Bb * HID;
  float* c2 = ws + off; off += (size_t)Bb * HID;
  int* scnt = (int*)(ws + off); off += 2;
  int* sgen = scnt + 1;

  init_sync_kernel<<<1, 32, 0, stream>>>(scnt, sgen);
  // 32768x1024x1024 GEMM: 32768 waves (4 tiles each) -> 4096 blocks
  xw0_kernel<<<4096, 256, 0, stream>>>(x, Wih0, bih0, bhh0, xw0);
  lstm_seq_kernel<<<PWG, PTHR, 0, stream>>>(xw0, Whh0, Wih1, Whh1, bih1, bhh1,
                                            Wih2, Whh2, bih2, bhh2,
                                            h0, h1, c0, c1, c2,
                                            h2all, scnt, sgen);
  // 32768x128x256 GEMM: 4096 waves (4 tiles each) -> 512 blocks
  fc1_kernel<<<512, 256, 0, stream>>>(h2all, fc1w, fc1b, r1);
  fc2_kernel<<<Bb, 256, 0, stream>>>(r1, fc2w, fc2b, (float*)d_out);
}